// GraphChannelMixerPyG_16045997818328
// MI455X (gfx1250) — compile-verified
//
#include <hip/hip_runtime.h>
#include <hip/hip_bf16.h>
#include <math.h>

#define ALPHA_C 0.05f
#define BETA_C  0.475f     // (1-alpha)/K_HOPS
#define NN 19              // nodes per graph
#define DD 64              // feature dim
#define KEIG 16
#define DIN 80             // DD + KEIG
#define TT 1024
#define BB 16
#define LN_EPS_C 1e-5f
#define SWEEPS 8

typedef __attribute__((ext_vector_type(2))) float v2f;
typedef __attribute__((ext_vector_type(8))) float v8f;
typedef __attribute__((ext_vector_type(4))) unsigned int u32x4;
typedef __attribute__((ext_vector_type(8))) int i32x8;
typedef __attribute__((ext_vector_type(4))) int i32x4;

// ---------------- Tensor Data Mover: 2-D fp32 tile descriptor (ISA ch.8) ----
// Moves rows of `width` fp32 elements, `rows` of them, global row stride
// `row_stride` elements. Optional LDS padding (load only): after every
// (1<<pad_int_code) * 2 DWORDs, skip (pad_amt_code+1) DWORDs in LDS.
template <bool IS_STORE>
__device__ __forceinline__ void tdm_2d(const void* gaddr, unsigned lds_byte_addr,
                                       unsigned width, unsigned rows,
                                       unsigned row_stride,
                                       bool pad, unsigned pad_int_code,
                                       unsigned pad_amt_code) {
  unsigned long long ga = (unsigned long long)gaddr;
  u32x4 g0;
  g0.x = 1u;                                        // count=1, is_restore=0
  g0.y = lds_byte_addr;                             // lds_addr
  g0.z = (unsigned)(ga & 0xFFFFFFFFu);              // global_addr[31:0]
  g0.w = (unsigned)((ga >> 32) & 0x01FFFFFFu)       // global_addr[56:32]
       | (2u << 30);                                // type = 2 ("image")
  i32x8 g1;
  unsigned w0 = (2u << 16);                         // data_size = 2 -> 4 bytes
  if (pad) w0 |= (1u << 20) | (pad_int_code << 22) | (pad_amt_code << 25);
  g1[0] = (int)w0;
  g1[1] = (int)((width & 0xFFFFu) << 16);                              // tensor_dim0 lo16 @ [63:48]
  g1[2] = (int)(((width >> 16) & 0xFFFFu) | ((rows & 0xFFFFu) << 16)); // dim0 hi | tensor_dim1 lo
  g1[3] = (int)(((rows >> 16) & 0xFFFFu) | ((width & 0xFFFFu) << 16)); // dim1 hi | tile_dim0
  g1[4] = (int)(rows & 0xFFFFu);                                       // tile_dim1 (tile_dim2=0)
  g1[5] = (int)row_stride;                                             // tensor_dim0_stride lo32
  g1[6] = 0;                                                           // stride hi | dim1_stride lo
  g1[7] = 0;
  i32x4 z4 = {0, 0, 0, 0};
  i32x8 z8 = {0, 0, 0, 0, 0, 0, 0, 0};
  if (IS_STORE) __builtin_amdgcn_tensor_store_from_lds(g0, g1, z4, z4, z8, 0);
  else          __builtin_amdgcn_tensor_load_to_lds(g0, g1, z4, z4, z8, 0);
}

// C(= Mt*16 x Nt*16) = A(lda-strided, zero-padded) * B (row stride rsB, col stride csB)
// via V_WMMA_F32_16X16X4_F32 tiles. All padding rows/cols of A/B must be zero.
__device__ __forceinline__ void wmma_gemm(const float* A, int lda,
                                          const float* B, int rsB, int csB,
                                          float* Out, int ldo,
                                          int Mt, int Nt, int K) {
  const int lane = threadIdx.x & 31;
  const int half = lane >> 4;      // 0: lanes 0-15, 1: lanes 16-31
  const int lm   = lane & 15;
  for (int mt = 0; mt < Mt; ++mt) {
    for (int nt = 0; nt < Nt; ++nt) {
      v8f c = {};
      for (int k0 = 0; k0 < K; k0 += 4) {
        v2f av, bv;
        // A 16x4 tile: lanes 0-15 hold K=k0,k0+1 ; lanes 16-31 hold K=k0+2,k0+3
        av.x = A[(mt * 16 + lm) * lda + (k0 + 2 * half + 0)];
        av.y = A[(mt * 16 + lm) * lda + (k0 + 2 * half + 1)];
        // B 4x16 tile: lanes 0-15 rows k0,k0+1 ; lanes 16-31 rows k0+2,k0+3
        bv.x = B[(k0 + 2 * half + 0) * rsB + (nt * 16 + lm) * csB];
        bv.y = B[(k0 + 2 * half + 1) * rsB + (nt * 16 + lm) * csB];
        c = __builtin_amdgcn_wmma_f32_16x16x4_f32(false, av, false, bv,
                                                  (short)0, c, false, false);
      }
      // D tile: VGPR r -> row r (lanes 0-15) / row r+8 (lanes 16-31)
#pragma unroll
      for (int r = 0; r < 8; ++r)
        Out[(mt * 16 + r + 8 * half) * ldo + nt * 16 + lm] = c[r];
    }
  }
}

__global__ __launch_bounds__(32)
void graph_mixer_fused(const float* __restrict__ feat,
                       const float* __restrict__ adj,
                       const float* __restrict__ p_ew,
                       const float* __restrict__ p_eb,
                       const float* __restrict__ g0W,
                       const float* __restrict__ g0b,
                       const float* __restrict__ g1W,
                       const float* __restrict__ g1b,
                       const float* __restrict__ ln0g,
                       const float* __restrict__ ln0b,
                       const float* __restrict__ ln1g,
                       const float* __restrict__ ln1b,
                       float* __restrict__ out) {
  __shared__ float aT[32 * 32];    // a_norm^T, zero padded (WMMA A-operand)
  __shared__ float Xa[32 * 80];    // X0 staging; TDM target for feature tile
  __shared__ float Xb[32 * 80];
  __shared__ float Hb[32 * 80];    // also used as raw-adjacency staging early on
  __shared__ float Xc[32 * 64];    // TDM source for the output tile
  __shared__ float S [20 * 20];    // Jacobi matrix; later reused to hold PE (19x16)
  __shared__ float V [20 * 20];    // eigenvectors; also early scratch for edge weights
  __shared__ float cA[16], sA[16];
  __shared__ float degL[32];

  const int g    = blockIdx.x;           // graph id in [0, B*T)
  const int b    = g / TT;
  const int t    = g % TT;
  const int lane = threadIdx.x;
  const float ew = p_ew[0];
  const float eb = p_eb[0];
  const float* adjG = adj + (size_t)g * NN * NN;
  const float* fG   = feat + (((size_t)b * NN) * TT + t) * DD;
  float*       outG = out  + (((size_t)b * NN) * TT + t) * DD;

  // ---- kick off TDM: feature tile (19 rows x 64 f32, global row stride T*D)
  // straight into Xa with LDS padding 64DW data + 16DW skip -> stride 80.
  // Runs fully overlapped with gcn_norm + Jacobi below (TENSORcnt-tracked).
  {
    unsigned ldsXa = (unsigned)(unsigned long long)(const void*)Xa;
    tdm_2d<false>(fG, ldsXa, DD, NN, TT * DD,
                  /*pad=*/true, /*interval 64 DW=*/5u, /*amount 16 DW=*/15u);
  }

  // ---- stage raw adjacency into Hb (stride 20) ----
  for (int i = lane; i < NN * NN; i += 32)
    Hb[(i / NN) * 20 + (i % NN)] = adjG[i];
  for (int i = lane; i < 32 * 32; i += 32) aT[i] = 0.f;
  __syncthreads();

  // ---- gcn_norm: w = softplus(adj*ew+eb)*mask (+ missing self loops) into V ----
  for (int idx = lane; idx < NN * NN; idx += 32) {
    int i = idx / NN, j = idx % NN;
    float a = Hb[i * 20 + j];
    float m = (a > 0.f) ? 1.f : 0.f;
    float x = a * ew + eb;
    float sp = (x > 20.f) ? x : log1pf(expf(x));
    float wv = sp * m;
    if (i == j && m == 0.f) wv = 1.f;       // add_remaining_self_loops
    V[i * 20 + j] = wv;
  }
  __syncthreads();
  if (lane < NN) {                           // deg over source axis (rows) per column
    float d = 0.f;
    for (int i = 0; i < NN; ++i) d += V[i * 20 + lane];
    degL[lane] = (d > 0.f) ? rsqrtf(fmaxf(d, 1e-30f)) : 0.f;
  }
  __syncthreads();
  // aT[m][k] = a_norm[k][m] = dis[k] * w[k][m] * dis[m]  (propagation is A^T x)
  for (int idx = lane; idx < NN * NN; idx += 32) {
    int m = idx / NN, k = idx % NN;
    aT[m * 32 + k] = degL[k] * V[k * 20 + m] * degL[m];
  }
  __syncthreads();

  // ---- Laplacian for dynamic PE: L = (1+1e-5) I - D^-1/2 A D^-1/2 ----
  if (lane < NN) {
    float d = 0.f;
    for (int j = 0; j < NN; ++j) d += Hb[lane * 20 + j];   // row sums
    degL[lane] = rsqrtf(fmaxf(d, 1e-6f));
  }
  __syncthreads();
  for (int idx = lane; idx < 400; idx += 32) {
    int i = idx / 20, j = idx % 20;
    float v = 0.f;
    if (i < NN && j < NN) {
      v = -degL[i] * Hb[i * 20 + j] * degL[j];
      if (i == j) v += 1.f + 1e-5f;
    }
    S[idx] = v;
    V[idx] = (i == j && i < NN) ? 1.f : 0.f;
  }
  __syncthreads();

  // ---- parallel-order cyclic Jacobi (round-robin tournament, 9 disjoint pairs) ----
  for (int sw = 0; sw < SWEEPS; ++sw) {
    for (int r = 0; r < 19; ++r) {
      if (lane >= 1 && lane <= 9) {
        int p = (r + lane) % 19, q = (r + 19 - lane) % 19;
        if (p > q) { int tw = p; p = q; q = tw; }
        float app = S[p * 20 + p], aqq = S[q * 20 + q], apq = S[p * 20 + q];
        float cc = 1.f, ss = 0.f;
        if (fabsf(apq) > 1e-12f) {
          float tau = (aqq - app) / (2.f * apq);
          float tt = ((tau >= 0.f) ? 1.f : -1.f) / (fabsf(tau) + sqrtf(1.f + tau * tau));
          cc = rsqrtf(1.f + tt * tt);
          ss = tt * cc;
        }
        cA[lane] = cc; sA[lane] = ss;
      }
      __syncthreads();
      // row phase: rows of each pair are disjoint across pairs
      for (int idx = lane; idx < 9 * 19; idx += 32) {
        int k = idx / 19 + 1, col = idx % 19;
        int p = (r + k) % 19, q = (r + 19 - k) % 19;
        if (p > q) { int tw = p; p = q; q = tw; }
        float cc = cA[k], ss = sA[k];
        float ap = S[p * 20 + col], aq = S[q * 20 + col];
        S[p * 20 + col] = cc * ap - ss * aq;
        S[q * 20 + col] = ss * ap + cc * aq;
      }
      __syncthreads();
      // column phase on S and V: columns disjoint across pairs
      for (int idx = lane; idx < 9 * 19; idx += 32) {
        int k = idx / 19 + 1, row = idx % 19;
        int p = (r + k) % 19, q = (r + 19 - k) % 19;
        if (p > q) { int tw = p; p = q; q = tw; }
        float cc = cA[k], ss = sA[k];
        float ap = S[row * 20 + p], aq = S[row * 20 + q];
        S[row * 20 + p] = cc * ap - ss * aq;
        S[row * 20 + q] = ss * ap + cc * aq;
        float vp = V[row * 20 + p], vq = V[row * 20 + q];
        V[row * 20 + p] = cc * vp - ss * vq;
        V[row * 20 + q] = ss * vp + cc * vq;
      }
      __syncthreads();
    }
  }

  // ---- select 16 smallest eigenvalues, sign fix, nan_to_num; PE -> S[0..303] ----
  int myrank = 32;
  float sgn = 1.f;
  if (lane < NN) {
    float lj = S[lane * 20 + lane];
    int rk = 0;
    for (int i = 0; i < NN; ++i) {
      float li = S[i * 20 + i];
      rk += (li < lj || (li == lj && i < lane)) ? 1 : 0;
    }
    myrank = rk;
    float ssum = 0.f;
    for (int i = 0; i < NN; ++i) ssum += V[i * 20 + lane];
    sgn = (ssum >= 0.f) ? 1.f : -1.f;        // sign(0) -> 1
  }
  __syncthreads();
  if (lane < NN && myrank < KEIG) {
    for (int i = 0; i < NN; ++i) {
      float v = V[i * 20 + lane] * sgn;
      if (v != v) v = 0.f;
      else if (fabsf(v) > 3.0e38f) v = (v > 0.f) ? 1.f : -1.f;
      S[i * KEIG + myrank] = v;
    }
  }
  __syncthreads();

  // ---- zero conv buffers (padding rows must be zero for WMMA operands).
  // Xa rows<19 belong to the in-flight TDM load (cols 0..63) + PE (cols 64..79);
  // only its padding rows 19..31 are zeroed here.
  for (int i = lane + NN * 80; i < 32 * 80; i += 32) Xa[i] = 0.f;
  for (int i = lane; i < 32 * 80; i += 32) { Xb[i] = 0.f; Hb[i] = 0.f; }
  for (int i = lane; i < 32 * 64; i += 32) Xc[i] = 0.f;
  // PE columns of X0
  for (int idx = lane; idx < NN * KEIG; idx += 32) {
    int n = idx / KEIG, k = idx % KEIG;
    Xa[n * 80 + DD + k] = S[n * KEIG + k];
  }
  __syncthreads();
  __builtin_amdgcn_s_wait_tensorcnt(0);     // feature tile now resident in Xa

  // ================= conv stage 0 (K = 80 features) =================
  wmma_gemm(aT, 32, Xa, 80, 1, Xb, 80, 2, 5, 32);          // Xb = P X0
  __syncthreads();
  for (int i = lane; i < NN * 80; i += 32) {
    int n = i / 80, c = i % 80;
    Hb[n * 80 + c] = ALPHA_C * Xa[n * 80 + c] + BETA_C * Xb[n * 80 + c];
  }
  __syncthreads();
  wmma_gemm(aT, 32, Xb, 80, 1, Xa, 80, 2, 5, 32);          // Xa = P^2 X0
  __syncthreads();
  for (int i = lane; i < NN * 80; i += 32) {
    int n = i / 80, c = i % 80;
    Hb[n * 80 + c] += BETA_C * Xa[n * 80 + c];
  }
  __syncthreads();
  // Xb[:, :64] = H @ W0^T  (B[k][o] = g0W[o*80+k] read straight from global/L2)
  wmma_gemm(Hb, 80, g0W, 1, 80, Xb, 80, 2, 4, 80);
  __syncthreads();

  // ---- layernorm 0 (+bias), in place on Xb[:, :64] ----
  for (int n = 0; n < NN; ++n) {
    float x0 = Xb[n * 80 + lane]      + g0b[lane];
    float x1 = Xb[n * 80 + 32 + lane] + g0b[32 + lane];
    float sm = x0 + x1;
    for (int m = 16; m >= 1; m >>= 1) sm += __shfl_xor(sm, m, 32);
    float mu = sm * (1.f / 64.f);
    float d0 = x0 - mu, d1 = x1 - mu;
    float vs = d0 * d0 + d1 * d1;
    for (int m = 16; m >= 1; m >>= 1) vs += __shfl_xor(vs, m, 32);
    float inv = rsqrtf(vs * (1.f / 64.f) + LN_EPS_C);
    Xb[n * 80 + lane]      = d0 * inv * ln0g[lane]      + ln0b[lane];
    Xb[n * 80 + 32 + lane] = d1 * inv * ln0g[32 + lane] + ln0b[32 + lane];
  }
  for (int n = NN; n < 32; ++n) {  // keep padding rows of Xb zero in first 64 cols
    Xb[n * 80 + lane] = 0.f; Xb[n * 80 + 32 + lane] = 0.f;
  }
  __syncthreads();

  // ================= conv stage 1 (K = 64) + residual =================
  wmma_gemm(aT, 32, Xb, 80, 1, Xa, 80, 2, 4, 32);          // Xa = P h0
  __syncthreads();
  for (int i = lane; i < NN * DD; i += 32) {
    int n = i / DD, c = i % DD;
    Hb[n * 80 + c] = ALPHA_C * Xb[n * 80 + c] + BETA_C * Xa[n * 80 + c];
  }
  __syncthreads();
  wmma_gemm(aT, 32, Xa, 80, 1, Xc, 64, 2, 4, 32);          // Xc = P^2 h0
  __syncthreads();
  for (int i = lane; i < NN * DD; i += 32) {
    int n = i / DD, c = i % DD;
    Hb[n * 80 + c] += BETA_C * Xc[n * 64 + c];
  }
  __syncthreads();
  wmma_gemm(Hb, 80, g1W, 1, 64, Xc, 64, 2, 4, 64);         // Xc = H @ W1^T
  __syncthreads();

  // ---- residual + layernorm 1 into Xc (contiguous 19x64 tile) ----
  for (int n = 0; n < NN; ++n) {
    float x0 = Xc[n * 64 + lane]      + g1b[lane]      + Xb[n * 80 + lane];
    float x1 = Xc[n * 64 + 32 + lane] + g1b[32 + lane] + Xb[n * 80 + 32 + lane];
    float sm = x0 + x1;
    for (int m = 16; m >= 1; m >>= 1) sm += __shfl_xor(sm, m, 32);
    float mu = sm * (1.f / 64.f);
    float d0 = x0 - mu, d1 = x1 - mu;
    float vs = d0 * d0 + d1 * d1;
    for (int m = 16; m >= 1; m >>= 1) vs += __shfl_xor(vs, m, 32);
    float inv = rsqrtf(vs * (1.f / 64.f) + LN_EPS_C);
    Xc[n * 64 + lane]      = d0 * inv * ln1g[lane]      + ln1b[lane];
    Xc[n * 64 + 32 + lane] = d1 * inv * ln1g[32 + lane] + ln1b[32 + lane];
  }
  __syncthreads();   // drain DS stores before the TDM engine reads Xc

  // ---- TDM store: LDS-contiguous 19x64 tile -> out[b][:,t,:] (row stride T*D)
  {
    unsigned ldsXc = (unsigned)(unsigned long long)(const void*)Xc;
    tdm_2d<true>(outG, ldsXc, DD, NN, TT * DD,
                 /*pad=*/false, 0u, 0u);
  }
  __builtin_amdgcn_s_wait_tensorcnt(0);
}

extern "C" void kernel_launch(void* const* d_in, const int* in_sizes, int n_in,
                              void* d_out, int out_size, void* d_ws, size_t ws_size,
                              hipStream_t stream) {
  const float* feat = (const float*)d_in[0];
  const float* adjp = (const float*)d_in[1];
  const float* ew   = (const float*)d_in[2];
  const float* ebp  = (const float*)d_in[3];
  const float* g0W  = (const float*)d_in[4];
  const float* g0b  = (const float*)d_in[5];
  const float* g1W  = (const float*)d_in[6];
  const float* g1b  = (const float*)d_in[7];
  const float* l0g  = (const float*)d_in[8];
  const float* l0b  = (const float*)d_in[9];
  const float* l1g  = (const float*)d_in[10];
  const float* l1b  = (const float*)d_in[11];
  float* outp = (float*)d_out;
  (void)in_sizes; (void)n_in; (void)out_size; (void)d_ws; (void)ws_size;

  dim3 grid(BB * TT);   // one wave32 per graph
  dim3 block(32);
  graph_mixer_fused<<<grid, block, 0, stream>>>(feat, adjp, ew, ebp, g0W, g0b,
                                                g1W, g1b, l0g, l0b, l1g, l1b, outp);
}